// VectorQuantizerEMA_36490042147277
// MI455X (gfx1250) — compile-verified
//
#include <hip/hip_runtime.h>

// ---------------------------------------------------------------------------
// VQ-VAE EMA vector quantizer for MI455X (gfx1250, wave32, WMMA).
// Distance GEMM (65536x1024x128 fp32) on V_WMMA_F32_16X16X4_F32, fused with
// row-wise argmin; LDS ping-pong double buffering for the embedding tiles and
// two independent WMMA accumulation chains to keep the matrix pipe fed.
// ---------------------------------------------------------------------------

#define DECAY       0.99f
#define ONE_M_DECAY 0.01f
#define EPSF        1e-5f

constexpr int Bc = 16, Tc = 4096, Dc = 128, Kc = 1024;
constexpr int Nc = Bc * Tc;           // 65536 rows
constexpr int DP = 132;               // padded LDS row stride (dwords) -> bank-conflict-free
constexpr int NTILES = Kc / 16;       // 64 column tiles

typedef float v2f __attribute__((ext_vector_type(2)));
typedef float v8f __attribute__((ext_vector_type(8)));

// ---------------------------------------------------------------- enorm[K]
__global__ void __launch_bounds__(128)
vq_enorm_kernel(const float* __restrict__ E, float* __restrict__ enorm) {
    const int wave = threadIdx.x >> 5, lane = threadIdx.x & 31;
    const int row  = blockIdx.x * 4 + wave;
    const float4 v = *(const float4*)(E + (size_t)row * Dc + lane * 4);
    float s = v.x * v.x + v.y * v.y + v.z * v.z + v.w * v.w;
    #pragma unroll
    for (int off = 16; off > 0; off >>= 1) s += __shfl_xor(s, off, 32);
    if (lane == 0) enorm[row] = s;
}

// ------------------------------------------- fused distance GEMM + argmin
// block = 128 threads (4 waves); each wave owns 16 rows of z.
// grid  = N / 64 = 1024 blocks.
__global__ void __launch_bounds__(128)
vq_dist_argmin_kernel(const float* __restrict__ z, const float* __restrict__ E,
                      const float* __restrict__ enorm,
                      int* __restrict__ idx_out, float* __restrict__ idx_f_out,
                      float* __restrict__ loss_acc) {
    __shared__ float etile[2][16 * DP];   // ping-pong buffers, padded rows
    __shared__ float znorm_s[4][16];

    const int tid  = threadIdx.x;
    const int wave = tid >> 5, lane = tid & 31;
    const int half = lane >> 4;           // which 16-lane half
    const int m    = lane & 15;
    const int row0 = (blockIdx.x * 4 + wave) * 16;

    // --- load A tile (16x128 fp32) into registers in WMMA 16x16x4 A-layout.
    // frag f covers K = 4f..4f+3; lane holds (row m, k = 4f+2*half, +1).
    v2f a[32];
    float zn = 0.f;
    const float* zrow = z + (size_t)(row0 + m) * Dc;
    #pragma unroll
    for (int f = 0; f < 32; ++f) {
        const int kb = 4 * f + 2 * half;
        a[f] = *(const v2f*)(zrow + kb);
        zn  += a[f].x * a[f].x + a[f].y * a[f].y;
    }
    zn += __shfl_xor(zn, 16, 32);         // combine the two K-subsets
    if (half == 0) znorm_s[wave][m] = zn;

    // cooperative-stage addressing: each thread moves 16 floats (4x float4)
    int rr[4], cc[4];
    #pragma unroll
    for (int j = 0; j < 4; ++j) {
        const int q = j * 512 + tid * 4;
        rr[j] = q >> 7;
        cc[j] = q & 127;
    }

    // prologue: stage tile 0 into buffer 0
    #pragma unroll
    for (int j = 0; j < 4; ++j) {
        *(float4*)&etile[0][rr[j] * DP + cc[j]] =
            *(const float4*)(E + (size_t)rr[j] * Dc + cc[j]);
    }
    __syncthreads();

    float minv[8];
    int   mini[8];
    #pragma unroll
    for (int i = 0; i < 8; ++i) { minv[i] = 3.4e38f; mini[i] = 0; }

    for (int ct = 0; ct < NTILES; ++ct) {
        const int colBase = ct * 16;
        const int buf = ct & 1;

        // prefetch next embedding tile into registers (overlaps the WMMAs)
        float4 pf[4];
        if (ct + 1 < NTILES) {
            #pragma unroll
            for (int j = 0; j < 4; ++j)
                pf[j] = *(const float4*)(E + (size_t)(colBase + 16 + rr[j]) * Dc + cc[j]);
        }

        // two independent WMMA accumulation chains (even/odd fragments)
        v8f acc0 = {0.f, 0.f, 0.f, 0.f, 0.f, 0.f, 0.f, 0.f};
        v8f acc1 = {0.f, 0.f, 0.f, 0.f, 0.f, 0.f, 0.f, 0.f};
        const float* brow = &etile[buf][m * DP + 2 * half];
        #pragma unroll
        for (int f = 0; f < 32; f += 2) {
            const v2f b0 = *(const v2f*)(brow + 4 * f);
            const v2f b1 = *(const v2f*)(brow + 4 * f + 4);
            acc0 = __builtin_amdgcn_wmma_f32_16x16x4_f32(
                false, a[f],     false, b0, (short)0, acc0, false, false);
            acc1 = __builtin_amdgcn_wmma_f32_16x16x4_f32(
                false, a[f + 1], false, b1, (short)0, acc1, false, false);
        }

        // store prefetched tile into the other buffer, then one barrier
        if (ct + 1 < NTILES) {
            #pragma unroll
            for (int j = 0; j < 4; ++j)
                *(float4*)&etile[buf ^ 1][rr[j] * DP + cc[j]] = pf[j];
        }
        __syncthreads();

        // C layout: element i of lane -> row = i + 8*half, col = colBase + m
        const float en = enorm[colBase + m];
        #pragma unroll
        for (int i = 0; i < 8; ++i) {
            const float dp = acc0[i] + acc1[i];
            const float s  = fmaf(-2.f, dp, en);      // dist - ||z||^2
            if (s < minv[i]) { minv[i] = s; mini[i] = colBase + m; }
        }
    }

    // min/argmin reduce across the 16 lanes of each half (tie -> lower index)
    float lsum = 0.f;
    #pragma unroll
    for (int i = 0; i < 8; ++i) {
        float v  = minv[i];
        int   ix = mini[i];
        #pragma unroll
        for (int off = 1; off < 16; off <<= 1) {
            const float ov = __shfl_xor(v, off, 32);
            const int   oi = __shfl_xor(ix, off, 32);
            if (ov < v || (ov == v && oi < ix)) { v = ov; ix = oi; }
        }
        if (m == 0) {
            const int grow = row0 + i + 8 * half;
            idx_out[grow]   = ix;
            idx_f_out[grow] = (float)ix;
            lsum += v + znorm_s[wave][i + 8 * half];  // true ||z - e||^2
        }
    }
    if (m == 0) atomicAdd(loss_acc, lsum);
}

// ------------------------------------- gather z_q + atomic segment sums
// block = 256 (8 waves); one wave per z row; grid = N/8.
__global__ void __launch_bounds__(256)
vq_gather_scatter_kernel(const float* __restrict__ z, const float* __restrict__ E,
                         const int* __restrict__ idx,
                         float* __restrict__ zq_out,
                         float* __restrict__ batch_ema,
                         float* __restrict__ batch_cluster) {
    const int wave = threadIdx.x >> 5, lane = threadIdx.x & 31;
    const int n = blockIdx.x * 8 + wave;
    const int k = idx[n];
    const float4 e4 = *(const float4*)(E + (size_t)k * Dc + lane * 4);
    *(float4*)(zq_out + (size_t)n * Dc + lane * 4) = e4;
    const float4 z4 = *(const float4*)(z + (size_t)n * Dc + lane * 4);
    float* ep = batch_ema + (size_t)k * Dc + lane * 4;
    atomicAdd(ep + 0, z4.x);
    atomicAdd(ep + 1, z4.y);
    atomicAdd(ep + 2, z4.z);
    atomicAdd(ep + 3, z4.w);
    if (lane == 0) atomicAdd(batch_cluster + k, 1.0f);
}

// ---------------------------------------------- EMA update + normalization
// single block of K=1024 threads
__global__ void __launch_bounds__(1024)
vq_finalize_kernel(const float* __restrict__ cluster_size,
                   const float* __restrict__ ema_w,
                   const float* __restrict__ batch_cluster,
                   const float* __restrict__ batch_ema,
                   const float* __restrict__ loss_acc,
                   float* __restrict__ out_loss,
                   float* __restrict__ out_emb,
                   float* __restrict__ out_cs,
                   float* __restrict__ out_emaw) {
    __shared__ float red[1024];
    const int k = threadIdx.x;
    const float ncs = cluster_size[k] * DECAY + ONE_M_DECAY * batch_cluster[k];
    out_cs[k] = ncs;
    red[k] = ncs;
    __syncthreads();
    #pragma unroll
    for (int s = 512; s > 0; s >>= 1) {
        if (k < s) red[k] += red[k + s];
        __syncthreads();
    }
    const float nsum = red[0];
    const float smoothed = (ncs + EPSF) / (nsum + (float)Kc * EPSF) * nsum;
    const float inv = 1.0f / smoothed;
    for (int d = 0; d < Dc; ++d) {
        const size_t o = (size_t)k * Dc + d;
        const float nw = ema_w[o] * DECAY + ONE_M_DECAY * batch_ema[o];
        out_emaw[o] = nw;
        out_emb[o]  = nw * inv;
    }
    if (k == 0) out_loss[0] = loss_acc[0] / (float)((size_t)Nc * Dc);
}

// ---------------------------------------------------------------------------
extern "C" void kernel_launch(void* const* d_in, const int* in_sizes, int n_in,
                              void* d_out, int out_size, void* d_ws, size_t ws_size,
                              hipStream_t stream) {
    (void)in_sizes; (void)n_in; (void)out_size; (void)ws_size;

    const float* z   = (const float*)d_in[0];   // [N, D]
    const float* E   = (const float*)d_in[1];   // [K, D]
    const float* cs  = (const float*)d_in[2];   // [K]
    const float* emw = (const float*)d_in[3];   // [K, D]

    // d_out: z_q_st | indices(float) | loss | new_embedding | new_cluster | new_ema_w
    float* out      = (float*)d_out;
    float* out_zq   = out;
    float* out_idx  = out_zq + (size_t)Nc * Dc;
    float* out_loss = out_idx + Nc;
    float* out_emb  = out_loss + 1;
    float* out_cs   = out_emb + (size_t)Kc * Dc;
    float* out_emaw = out_cs + Kc;

    // workspace: [zeroed accumulators][idx][enorm]
    char*  ws        = (char*)d_ws;
    float* batch_ema = (float*)ws;                 ws += sizeof(float) * Kc * Dc;
    float* batch_cl  = (float*)ws;                 ws += sizeof(float) * Kc;
    float* loss_acc  = (float*)ws;                 ws += sizeof(float);
    int*   idx_i     = (int*)ws;                   ws += sizeof(int) * Nc;
    float* enorm     = (float*)ws;                 ws += sizeof(float) * Kc;

    hipMemsetAsync(batch_ema, 0, sizeof(float) * ((size_t)Kc * Dc + Kc + 1), stream);

    vq_enorm_kernel<<<Kc / 4, 128, 0, stream>>>(E, enorm);

    vq_dist_argmin_kernel<<<Nc / 64, 128, 0, stream>>>(
        z, E, enorm, idx_i, out_idx, loss_acc);

    vq_gather_scatter_kernel<<<Nc / 8, 256, 0, stream>>>(
        z, E, idx_i, out_zq, batch_ema, batch_cl);

    vq_finalize_kernel<<<1, 1024, 0, stream>>>(
        cs, emw, batch_cl, batch_ema, loss_acc,
        out_loss, out_emb, out_cs, out_emaw);
}